// TDGNN_GraphSage_30099130811051
// MI455X (gfx1250) — compile-verified
//
#include <hip/hip_runtime.h>

// ---------------------------------------------------------------------------
// GraphSage 2-layer encoder, restructured for MI455X (gfx1250, wave32, WMMA).
//
//   out[e,c] = 0.05 * sum_{r in [20e,20e+20)} sum_d relu(h1[r,d]) * Wcomb[c,d]
//   h1[r,:]  = (0.1 * sum_j features[nidx[node2(r)*K + j], :]) @ W1^T
//   Wcomb    = Wc @ W2                           (folds layer-2 + classifier)
//
// Feature table (102 MB) is L2-resident on MI455X (192 MB L2) -> gather-bound,
// so the GEMM runs in native FP32 via V_WMMA_F32_16X16X4_F32 (exact numerics,
// zero roofline cost).
// ---------------------------------------------------------------------------

typedef __attribute__((ext_vector_type(2))) float v2f;
typedef __attribute__((ext_vector_type(8))) float v8f;

#define EDGES   4096
#define FEAT    256
#define EMBED   128
#define KNEI    10
#define ROWS    (EDGES * 2 * KNEI)   // 81920 layer-1 rows
#define MTILE   32                   // rows per workgroup
#define F1_PAD  260                  // 256 + 4 : conflict-free WMMA A reads
#define H1_PAD  132                  // 128 + 4 : conflict-light epilogue reads

// ---------------------------------------------------------------------------
// Prep: Wcomb = Wc @ W2 (2x128) into workspace; zero the output accumulator.
// ---------------------------------------------------------------------------
__global__ __launch_bounds__(256) void sage_prep_kernel(
    const float* __restrict__ W2,   // [128,128]
    const float* __restrict__ Wc,   // [2,128]
    float* __restrict__ wcomb,      // [2,128] (workspace)
    float* __restrict__ out)        // [4096,2]
{
    const int t = threadIdx.x;
    const int c = t >> 7;           // 0..1
    const int d = t & 127;          // 0..127
    float s = 0.0f;
#pragma unroll 8
    for (int g = 0; g < EMBED; ++g)
        s += Wc[c * EMBED + g] * W2[g * EMBED + d];
    wcomb[c * EMBED + d] = s;

    for (int i = t; i < EDGES * 2; i += 256)
        out[i] = 0.0f;
}

// ---------------------------------------------------------------------------
// Fused gather -> mean -> FP32 WMMA GEMM -> relu -> group-reduce -> atomicAdd.
// 256 threads = 8 waves; one block handles 32 layer-1 rows.
// ---------------------------------------------------------------------------
__global__ __launch_bounds__(256) void sage_main_kernel(
    const int*   __restrict__ edges,   // [E,2]
    const int*   __restrict__ nidx,    // [N,K]
    const float* __restrict__ feat,    // [N,FEAT]
    const float* __restrict__ W1,      // [EMBED,FEAT]
    const float* __restrict__ wcomb,   // [2,EMBED]
    float*       __restrict__ out)     // [E,2]
{
    __shared__ float f1[MTILE][F1_PAD];   // aggregated features (A matrix)
    __shared__ float h1[MTILE][H1_PAD];   // relu(GEMM) tile

    const int t    = threadIdx.x;
    const int row0 = blockIdx.x * MTILE;

    // -------- Phase 1: gather + mean over 2-hop neighbors (0.1 * sum_j) ----
    {
        const int lrow = t >> 3;                 // 0..31 local row
        const int part = t & 7;                  // 8 threads per row
        const int r    = row0 + lrow;            // global layer-1 row
        const int e    = r / 20;
        const int p    = (r / 10) & 1;
        const int k    = r % 10;
        const int node2 = nidx[(size_t)edges[e * 2 + p] * KNEI + k];
        const int f0    = part * 32;             // 32 features per thread

        float acc[32];
#pragma unroll
        for (int i = 0; i < 32; ++i) acc[i] = 0.0f;

        for (int j = 0; j < KNEI; ++j) {
            const int node1 = nidx[(size_t)node2 * KNEI + j];
            const float4* __restrict__ src =
                (const float4*)(feat + (size_t)node1 * FEAT + f0);
#pragma unroll
            for (int q = 0; q < 8; ++q) {
                const float4 v = src[q];
                acc[q * 4 + 0] += v.x;
                acc[q * 4 + 1] += v.y;
                acc[q * 4 + 2] += v.z;
                acc[q * 4 + 3] += v.w;
            }
        }
#pragma unroll
        for (int i = 0; i < 32; ++i)
            f1[lrow][f0 + i] = acc[i] * 0.1f;    // mean over K=10
    }
    __syncthreads();

    // -------- Phase 2: FP32 WMMA GEMM  h1 = f1 @ W1^T ----------------------
    // Wave w owns output columns [16w, 16w+16); two 16-row M-subtiles.
    // A 16x4 f32 layout: lane L -> row (L&15), K-half (L>>4)*2.
    // B  4x16 f32 layout mirrors it with N = (L&15), so B rows index W1[n][k].
    const int wave   = t >> 5;
    const int lane   = t & 31;
    const int lrow16 = lane & 15;
    const int khalf  = (lane >> 4) * 2;

    v8f acc0 = {};   // M-subtile 0 (rows 0..15)
    v8f acc1 = {};   // M-subtile 1 (rows 16..31)

    const float* __restrict__ w1row =
        W1 + (size_t)(wave * 16 + lrow16) * FEAT + khalf;
    const float* __restrict__ a0p = &f1[lrow16][khalf];
    const float* __restrict__ a1p = &f1[16 + lrow16][khalf];

    for (int k0 = 0; k0 < FEAT; k0 += 4) {
        v2f b;  b.x  = w1row[k0]; b.y  = w1row[k0 + 1];   // L2-resident W1
        v2f a0; a0.x = a0p[k0];   a0.y = a0p[k0 + 1];     // LDS, bank-clean
        v2f a1; a1.x = a1p[k0];   a1.y = a1p[k0 + 1];
        acc0 = __builtin_amdgcn_wmma_f32_16x16x4_f32(
                   false, a0, false, b, (short)0, acc0, false, false);
        acc1 = __builtin_amdgcn_wmma_f32_16x16x4_f32(
                   false, a1, false, b, (short)0, acc1, false, false);
    }

    // -------- Phase 3: relu -> LDS (C layout: VGPR i, lane L -> M = i+8*(L>>4),
    //                                N = (L&15) + 16*wave) --------------------
    {
        const int n = wave * 16 + lrow16;
        const int mhi = 8 * (lane >> 4);
#pragma unroll
        for (int i = 0; i < 8; ++i) {
            h1[mhi + i][n]      = fmaxf(acc0[i], 0.0f);
            h1[16 + mhi + i][n] = fmaxf(acc1[i], 0.0f);
        }
    }
    __syncthreads();

    // -------- Phase 4: per-row dot with Wcomb, atomic accumulate -----------
    if (t < 64) {
        const int m = t & 31;                    // local row
        const int c = t >> 5;                    // class 0/1
        const float* __restrict__ wc = wcomb + c * EMBED;
        float s = 0.0f;
#pragma unroll 8
        for (int n = 0; n < EMBED; ++n)
            s += h1[m][n] * wc[n];
        const int r = row0 + m;
        atomicAdd(out + (r / 20) * 2 + c, 0.05f * s);
    }
}

// ---------------------------------------------------------------------------
extern "C" void kernel_launch(void* const* d_in, const int* in_sizes, int n_in,
                              void* d_out, int out_size, void* d_ws, size_t ws_size,
                              hipStream_t stream)
{
    const int*   edges = (const int*)  d_in[0];   // [4096,2]
    const int*   nidx  = (const int*)  d_in[1];   // [100000,10]
    const float* feat  = (const float*)d_in[2];   // [100000,256]
    const float* W1    = (const float*)d_in[3];   // [128,256]
    const float* W2    = (const float*)d_in[4];   // [128,128]
    const float* Wc    = (const float*)d_in[5];   // [2,128]
    float*       out   = (float*)d_out;           // [4096,2]
    float*       wcomb = (float*)d_ws;            // 2*128 floats

    sage_prep_kernel<<<1, 256, 0, stream>>>(W2, Wc, wcomb, out);

    const int nblocks = ROWS / MTILE;             // 2560
    sage_main_kernel<<<nblocks, 256, 0, stream>>>(edges, nidx, feat, W1, wcomb, out);
}